// PoseEstimator_40157944218247
// MI455X (gfx1250) — compile-verified
//
#include <hip/hip_runtime.h>

typedef __attribute__((ext_vector_type(2))) float f32x2;
typedef __attribute__((ext_vector_type(8))) float f32x8;

#define BPB 128   // blocks per batch
#define TPB 256   // 8 wave32 waves

// ---------------------------------------------------------------------------
// Per-batch constant folding:  P = R_l * K_l^-1,  M = K_sat * R_sat^T,
// Q = M * P,  c0 = M * t.   Also zeroes the per-batch accumulators.
// ws layout (floats): [b*24 .. b*24+20] = M(9),Q(9),c0(3); acc at B*24 + b*16.
// ---------------------------------------------------------------------------
__global__ void pose_setup_kernel(const float* __restrict__ K_left,
                                  const float* __restrict__ R_l,
                                  const float* __restrict__ K_sat,
                                  const float* __restrict__ R_sat,
                                  const float* __restrict__ t_sat,
                                  float* __restrict__ ws, int B)
{
    int b = threadIdx.x;
    if (b >= B) return;
    const float* Kl = K_left + b * 9;
    const float* Rl = R_l    + b * 9;
    const float* Ks = K_sat  + b * 9;
    const float* Rs = R_sat  + b * 9;
    const float* tv = t_sat  + b * 3;

    // general 3x3 inverse of K_left via adjugate
    float a00=Kl[0],a01=Kl[1],a02=Kl[2],a10=Kl[3],a11=Kl[4],a12=Kl[5],a20=Kl[6],a21=Kl[7],a22=Kl[8];
    float det = a00*(a11*a22-a12*a21) - a01*(a10*a22-a12*a20) + a02*(a10*a21-a11*a20);
    float id  = 1.0f / det;
    float Ki[9];
    Ki[0]=(a11*a22-a12*a21)*id; Ki[1]=(a02*a21-a01*a22)*id; Ki[2]=(a01*a12-a02*a11)*id;
    Ki[3]=(a12*a20-a10*a22)*id; Ki[4]=(a00*a22-a02*a20)*id; Ki[5]=(a02*a10-a00*a12)*id;
    Ki[6]=(a10*a21-a11*a20)*id; Ki[7]=(a01*a20-a00*a21)*id; Ki[8]=(a00*a11-a01*a10)*id;

    float P[9], M[9], Q[9];
#pragma unroll
    for (int i = 0; i < 3; ++i)
#pragma unroll
        for (int j = 0; j < 3; ++j)
            P[i*3+j] = Rl[i*3+0]*Ki[0*3+j] + Rl[i*3+1]*Ki[1*3+j] + Rl[i*3+2]*Ki[2*3+j];
#pragma unroll
    for (int i = 0; i < 3; ++i)
#pragma unroll
        for (int j = 0; j < 3; ++j)   // M = K_sat * R_sat^T
            M[i*3+j] = Ks[i*3+0]*Rs[j*3+0] + Ks[i*3+1]*Rs[j*3+1] + Ks[i*3+2]*Rs[j*3+2];
#pragma unroll
    for (int i = 0; i < 3; ++i)
#pragma unroll
        for (int j = 0; j < 3; ++j)
            Q[i*3+j] = M[i*3+0]*P[0*3+j] + M[i*3+1]*P[1*3+j] + M[i*3+2]*P[2*3+j];

    float* dst = ws + b * 24;
#pragma unroll
    for (int k = 0; k < 9; ++k) dst[k]     = M[k];
#pragma unroll
    for (int k = 0; k < 9; ++k) dst[9 + k] = Q[k];
#pragma unroll
    for (int i = 0; i < 3; ++i)
        dst[18 + i] = M[i*3+0]*tv[0] + M[i*3+1]*tv[1] + M[i*3+2]*tv[2];
    dst[21] = 0.f; dst[22] = 0.f; dst[23] = 0.f;

    float* acc = ws + B * 24 + b * 16;
#pragma unroll
    for (int k = 0; k < 16; ++k) acc[k] = 0.f;
}

// ---------------------------------------------------------------------------
// Streaming normal-equation accumulation (bandwidth-bound: 24 B/point).
// Block tree reduce uses V_WMMA_F32_16X16X4_F32 with an all-ones A matrix.
// ---------------------------------------------------------------------------
__global__ void __launch_bounds__(TPB)
pose_reduce_kernel(const float* __restrict__ x_left,
                   const float* __restrict__ depth,
                   const float* __restrict__ x_sat,
                   const float* __restrict__ weights,
                   float* __restrict__ ws, int N, int B)
{
    const int b = blockIdx.y;
    const float* cst = ws + b * 24;
    const float M00=cst[0],M01=cst[1],M02=cst[2],M10=cst[3],M11=cst[4],M12=cst[5],
                M20=cst[6],M21=cst[7],M22=cst[8];
    const float Q00=cst[9],Q01=cst[10],Q02=cst[11],Q10=cst[12],Q11=cst[13],Q12=cst[14],
                Q20=cst[15],Q21=cst[16],Q22=cst[17];
    const float c0x=cst[18],c0y=cst[19],c0z=cst[20];

    const float4* xl4 = reinterpret_cast<const float4*>(x_left + (size_t)b * N * 2);
    const float2* dd2 = reinterpret_cast<const float2*>(depth  + (size_t)b * N);
    const float4* xs4 = reinterpret_cast<const float4*>(x_sat  + (size_t)b * N * 2);
    const float2* ww2 = reinterpret_cast<const float2*>(weights + (size_t)b * N);

    float s00=0.f,s01=0.f,s02=0.f,s11=0.f,s12=0.f,s22=0.f,t0=0.f,t1=0.f,t2=0.f;

    auto accum = [&](float ul, float vl, float d, float us, float vs, float w) {
        float Cx = d * (Q00*ul + Q01*vl + Q02) - c0x;
        float Cy = d * (Q10*ul + Q11*vl + Q12) - c0y;
        float Cz = d * (Q20*ul + Q21*vl + Q22) - c0z;
        float a1x = M00 - us*M20, a1y = M01 - us*M21, a1z = M02 - us*M22;
        float a2x = M10 - vs*M20, a2y = M11 - vs*M21, a2z = M12 - vs*M22;
        float r1 = us*Cz - Cx;
        float r2 = vs*Cz - Cy;
        float wq = fmaxf(w, 0.f) + 1e-8f;   // sqrt cancels in normal equations
        s00 += wq*(a1x*a1x + a2x*a2x);
        s01 += wq*(a1x*a1y + a2x*a2y);
        s02 += wq*(a1x*a1z + a2x*a2z);
        s11 += wq*(a1y*a1y + a2y*a2y);
        s12 += wq*(a1y*a1z + a2y*a2z);
        s22 += wq*(a1z*a1z + a2z*a2z);
        t0  += wq*(a1x*r1 + a2x*r2);
        t1  += wq*(a1y*r1 + a2y*r2);
        t2  += wq*(a1z*r1 + a2z*r2);
    };

    const int pairs  = N >> 1;
    const int stride = gridDim.x * blockDim.x;
    for (int p = blockIdx.x * blockDim.x + threadIdx.x; p < pairs; p += stride) {
        float4 xl = xl4[p];
        float2 dd = dd2[p];
        float4 xs = xs4[p];
        float2 wv = ww2[p];
        accum(xl.x, xl.y, dd.x, xs.x, xs.y, wv.x);
        accum(xl.z, xl.w, dd.y, xs.z, xs.w, wv.y);
    }

    // ---- wave32 shuffle reduction of the 9 partials ----
    float vals[9] = {s00,s01,s02,s11,s12,s22,t0,t1,t2};
#pragma unroll
    for (int off = 16; off > 0; off >>= 1) {
#pragma unroll
        for (int j = 0; j < 9; ++j)
            vals[j] += __shfl_down(vals[j], off, 32);
    }

    __shared__ float sm[8 * 16];           // 8 waves x 16 padded columns
    const int wave = threadIdx.x >> 5;
    const int lane = threadIdx.x & 31;
    if (lane == 0) {
#pragma unroll
        for (int j = 0; j < 9; ++j)  sm[wave * 16 + j] = vals[j];
#pragma unroll
        for (int j = 9; j < 16; ++j) sm[wave * 16 + j] = 0.f;
    }
    __syncthreads();

    // ---- cross-wave reduction via WMMA: D[0][j] = sum over 8 waves of col j.
    // A = all-ones 16x4 (layout-invariant), B rows = wave partials (N = lane%16).
    // Every wave executes the WMMA (EXEC must be all-ones); result identical.
    const int col = lane & 15;
    const int hi  = lane >> 4;
    f32x2 aone; aone[0] = 1.f; aone[1] = 1.f;
    f32x2 b0, b1;
    b0[0] = sm[(2*hi + 0) * 16 + col];
    b0[1] = sm[(2*hi + 1) * 16 + col];
    b1[0] = sm[(2*hi + 4) * 16 + col];
    b1[1] = sm[(2*hi + 5) * 16 + col];
    f32x8 c = {};
    c = __builtin_amdgcn_wmma_f32_16x16x4_f32(false, aone, false, b0, (short)0, c, false, false);
    c = __builtin_amdgcn_wmma_f32_16x16x4_f32(false, aone, false, b1, (short)0, c, false, false);

    if (threadIdx.x < 9)                    // D row 0 lives in VGPR0, lanes 0..15
        atomicAdd(ws + B * 24 + b * 16 + threadIdx.x, c[0]);
}

// ---------------------------------------------------------------------------
// Tiny per-batch symmetric 3x3 solve (adjugate / Cramer).
// ---------------------------------------------------------------------------
__global__ void pose_solve_kernel(const float* __restrict__ ws,
                                  float* __restrict__ out, int B)
{
    int b = threadIdx.x;
    if (b >= B) return;
    const float* acc = ws + B * 24 + b * 16;
    float a=acc[0], bb=acc[1], c=acc[2], d=acc[3], e=acc[4], f=acc[5];
    float t0=acc[6], t1=acc[7], t2=acc[8];
    float det = a*(d*f - e*e) - bb*(bb*f - e*c) + c*(bb*e - d*c);
    float id  = 1.0f / det;
    float i00 = (d*f - e*e)  * id;
    float i01 = (c*e - bb*f) * id;
    float i02 = (bb*e - c*d) * id;
    float i11 = (a*f - c*c)  * id;
    float i12 = (bb*c - a*e) * id;
    float i22 = (a*d - bb*bb)* id;
    out[b*3 + 0] = i00*t0 + i01*t1 + i02*t2;
    out[b*3 + 1] = i01*t0 + i11*t1 + i12*t2;
    out[b*3 + 2] = i02*t0 + i12*t1 + i22*t2;
}

extern "C" void kernel_launch(void* const* d_in, const int* in_sizes, int n_in,
                              void* d_out, int out_size, void* d_ws, size_t ws_size,
                              hipStream_t stream)
{
    const float* x_left  = (const float*)d_in[0];
    const float* depth   = (const float*)d_in[1];
    const float* K_left  = (const float*)d_in[2];
    const float* R_l     = (const float*)d_in[3];
    const float* x_sat   = (const float*)d_in[4];
    const float* K_sat   = (const float*)d_in[5];
    const float* R_sat   = (const float*)d_in[6];
    const float* t_sat   = (const float*)d_in[7];
    const float* weights = (const float*)d_in[8];
    float* out = (float*)d_out;
    float* ws  = (float*)d_ws;

    const int B = in_sizes[2] / 9;        // (B,3,3)
    const int N = in_sizes[1] / B;        // (B,N)
    const int smallThreads = ((B + 31) / 32) * 32;

    pose_setup_kernel<<<1, smallThreads, 0, stream>>>(K_left, R_l, K_sat, R_sat, t_sat, ws, B);

    dim3 grid(BPB, B);
    pose_reduce_kernel<<<grid, TPB, 0, stream>>>(x_left, depth, x_sat, weights, ws, N, B);

    pose_solve_kernel<<<1, smallThreads, 0, stream>>>(ws, out, B);
}